// Explore_82678120448825
// MI455X (gfx1250) — compile-verified
//
#include <hip/hip_runtime.h>
#include <hip/hip_bf16.h>

#define NN 200000   // nodes
#define NE 500000   // edges
#define NQ 100
#define NR 401
#define DD 128      // D
#define AA 64       // A
#define NPAIR (NR * NQ)

typedef __attribute__((ext_vector_type(2))) float v2f;
typedef __attribute__((ext_vector_type(8))) float v8f;

// ---------------------------------------------------------------------------
// Zero a float4-aligned region.
// ---------------------------------------------------------------------------
__global__ __launch_bounds__(256) void zero_kernel(float4* __restrict__ p, int n4) {
    int i = blockIdx.x * 256 + threadIdx.x;
    if (i < n4) {
        float4 z = {0.f, 0.f, 0.f, 0.f};
        p[i] = z;
    }
}

// ---------------------------------------------------------------------------
// Xp[pair, d] = rela[rel, d] * q_emb[q, d]   for pair = rel*NQ + q
// ---------------------------------------------------------------------------
__global__ __launch_bounds__(256) void pair_prod_kernel(
    const float* __restrict__ rela, const float* __restrict__ qemb,
    float* __restrict__ Xp) {
    int idx = blockIdx.x * 256 + threadIdx.x;
    if (idx >= NPAIR * DD) return;
    int pair = idx >> 7;          // /128
    int d    = idx & 127;
    int rel  = pair / NQ;
    int q    = pair - rel * NQ;
    Xp[idx] = rela[rel * DD + d] * qemb[q * DD + d];
}

// ---------------------------------------------------------------------------
// Out[M, Nc] = X[M, 128] @ W[Nc, 128]^T  (+ bias[Nc] if non-null)
// f32 WMMA 16x16x4. Block = 256 threads = 8 waves; each wave: 16 rows, all Nc cols.
// W staged in LDS k-major to keep B-fragment reads (lanes 0-15 consecutive cols)
// nearly bank-conflict free.
// ---------------------------------------------------------------------------
template <int NT>
__global__ __launch_bounds__(256) void gemm_xwt_wmma(
    const float* __restrict__ X, const float* __restrict__ W,
    const float* __restrict__ bias, float* __restrict__ Out, int M) {
    constexpr int K  = 128;
    constexpr int Nc = NT * 16;
    __shared__ float Wl[K * Nc];   // Wl[k*Nc + col] = W[col*K + k]

    // Stage W transposed into LDS (coalesced global reads).
    {
        const float4* Wv = (const float4*)W;
        for (int i = threadIdx.x; i < Nc * (K / 4); i += 256) {
            int col = i >> 5;      // i / (K/4)
            int kq  = i & 31;
            float4 v = Wv[i];
            int k0 = kq * 4;
            Wl[(k0 + 0) * Nc + col] = v.x;
            Wl[(k0 + 1) * Nc + col] = v.y;
            Wl[(k0 + 2) * Nc + col] = v.z;
            Wl[(k0 + 3) * Nc + col] = v.w;
        }
    }
    __syncthreads();

    const int wave    = threadIdx.x >> 5;
    const int lane    = threadIdx.x & 31;
    const int rowBase = (blockIdx.x * 8 + wave) * 16;
    const int m       = lane & 15;          // M index (A) / N index (B,C)
    const int khalf   = (lane >> 4) * 2;    // K sub-offset for A/B fragments

    // C accumulators, initialized with bias (constant per lane: N = lane&15).
    v8f c[NT];
#pragma unroll
    for (int t = 0; t < NT; ++t) {
        float bv = bias ? bias[t * 16 + m] : 0.f;
#pragma unroll
        for (int r = 0; r < 8; ++r) c[t][r] = bv;
    }

    // Clamp A row so EXEC stays all-1s around WMMA (guard only the stores).
    int arow = rowBase + m;
    if (arow >= M) arow = M - 1;
    const float* xrow = X + (long)arow * K;

    for (int kb = 0; kb < K; kb += 4) {
        int kk = kb + khalf;
        float2 av = *(const float2*)(xrow + kk);
        v2f a;
        a[0] = av.x;
        a[1] = av.y;
#pragma unroll
        for (int t = 0; t < NT; ++t) {
            int col = t * 16 + m;
            v2f b;
            b[0] = Wl[kk * Nc + col];
            b[1] = Wl[(kk + 1) * Nc + col];
            c[t] = __builtin_amdgcn_wmma_f32_16x16x4_f32(
                false, a, false, b, (short)0, c[t], false, false);
        }
    }

    // C layout: VGPR r, lane l -> row = r + 8*(l>>4), col = l&15.
    const int rhi = (lane >> 4) << 3;
#pragma unroll
    for (int r = 0; r < 8; ++r) {
        int row = rowBase + r + rhi;
        if (row < M) {
#pragma unroll
            for (int t = 0; t < NT; ++t)
                Out[(long)row * Nc + t * 16 + m] = c[t][r];
        }
    }
}

// ---------------------------------------------------------------------------
// Wave-per-edge: attention logit via gathered 64-float rows + shuffle
// reduction, then alpha * (hs*hr) scatter-added into agg with HW f32 atomics.
// ---------------------------------------------------------------------------
__global__ __launch_bounds__(256) void edge_kernel(
    const float* __restrict__ WsH,    // [NN, 64]  (includes Ws_b)
    const float* __restrict__ WrR,    // [NR, 64]
    const float* __restrict__ WqQ,    // [NQ, 64]
    const float* __restrict__ P,      // [NPAIR, 64]
    const float* __restrict__ hidden, // [NN, 128]
    const float* __restrict__ rela,   // [NR, 128]
    const float* __restrict__ wa_w,   // [64]
    const float* __restrict__ wa_b,   // [1]
    const int* __restrict__ eq, const int* __restrict__ erel,
    const int* __restrict__ esub, const int* __restrict__ eobj,
    float* __restrict__ agg) {
    int e = blockIdx.x * 8 + (threadIdx.x >> 5);
    if (e >= NE) return;
    const int lane = threadIdx.x & 31;

    int sub = esub[e];
    int rel = erel[e];
    int q   = eq[e];
    int obj = eobj[e];

    const float* hs = hidden + (long)sub * DD;
    const float* hr = rela + (long)rel * DD;
    __builtin_prefetch(hs + lane, 0, 0);   // global_prefetch_b8
    __builtin_prefetch(hr + lane, 0, 0);

    const float* ws = WsH + (long)sub * AA;
    const float* wr = WrR + (long)rel * AA;
    const float* wq = WqQ + (long)q * AA;
    const float* pp = P + (long)(rel * NQ + q) * AA;

    int a0 = lane, a1 = lane + 32;
    float s0 = ws[a0] + wr[a0] + wq[a0] + pp[a0];
    float s1 = ws[a1] + wr[a1] + wq[a1] + pp[a1];
    float part = wa_w[a0] * fmaxf(s0, 0.f) + wa_w[a1] * fmaxf(s1, 0.f);
#pragma unroll
    for (int off = 16; off > 0; off >>= 1) part += __shfl_xor(part, off, 32);

    float logit = part + wa_b[0];
    float alpha = 1.f / (1.f + __expf(-logit));

    float* ag = agg + (long)obj * DD;
#pragma unroll
    for (int j = 0; j < 4; ++j) {
        int d = lane + j * 32;
        unsafeAtomicAdd(&ag[d], alpha * hs[d] * hr[d]);
    }
}

// ---------------------------------------------------------------------------
extern "C" void kernel_launch(void* const* d_in, const int* in_sizes, int n_in,
                              void* d_out, int out_size, void* d_ws, size_t ws_size,
                              hipStream_t stream) {
    const float* hidden = (const float*)d_in[0];
    const float* q_emb  = (const float*)d_in[1];
    const float* rela   = (const float*)d_in[2];
    const float* Ws_w   = (const float*)d_in[3];
    const float* Ws_b   = (const float*)d_in[4];
    const float* Wr_w   = (const float*)d_in[5];
    const float* Wq_w   = (const float*)d_in[6];
    const float* Wqr_w  = (const float*)d_in[7];
    const float* wa_w   = (const float*)d_in[8];
    const float* wa_b   = (const float*)d_in[9];
    const float* Wh_w   = (const float*)d_in[10];
    const int* edge_q   = (const int*)d_in[11];
    const int* edge_rel = (const int*)d_in[12];
    const int* edge_sub = (const int*)d_in[13];
    const int* edge_obj = (const int*)d_in[14];
    float* out = (float*)d_out;

    // Workspace layout (floats): agg | WsH | P | WrR | WqQ | Xp
    float* ws  = (float*)d_ws;
    float* agg = ws;
    float* WsH = agg + (size_t)NN * DD;
    float* P   = WsH + (size_t)NN * AA;
    float* WrR = P + (size_t)NPAIR * AA;
    float* WqQ = WrR + (size_t)NR * AA;
    float* Xp  = WqQ + (size_t)NQ * AA;

    // 1) zero agg
    {
        int n4 = NN * DD / 4;
        zero_kernel<<<(n4 + 255) / 256, 256, 0, stream>>>((float4*)agg, n4);
    }
    // 2) pair products Xp[rel*NQ+q, :] = hr * hq
    {
        int tot = NPAIR * DD;
        pair_prod_kernel<<<(tot + 255) / 256, 256, 0, stream>>>(rela, q_emb, Xp);
    }
    // 3) WMMA GEMMs: per-node and per-table attention pre-activations
    gemm_xwt_wmma<4><<<(NN + 127) / 128, 256, 0, stream>>>(hidden, Ws_w, Ws_b, WsH, NN);
    gemm_xwt_wmma<4><<<(NPAIR + 127) / 128, 256, 0, stream>>>(Xp, Wqr_w, nullptr, P, NPAIR);
    gemm_xwt_wmma<4><<<(NR + 127) / 128, 256, 0, stream>>>(rela, Wr_w, nullptr, WrR, NR);
    gemm_xwt_wmma<4><<<(NQ + 127) / 128, 256, 0, stream>>>(q_emb, Wq_w, nullptr, WqQ, NQ);
    // 4) edge attention + scatter-add (wave per edge)
    edge_kernel<<<(NE + 7) / 8, 256, 0, stream>>>(WsH, WrR, WqQ, P, hidden, rela,
                                                  wa_w, wa_b, edge_q, edge_rel,
                                                  edge_sub, edge_obj, agg);
    // 5) final projection: out = agg @ Wh^T  (WMMA, Nc = 128)
    gemm_xwt_wmma<8><<<(NN + 127) / 128, 256, 0, stream>>>(agg, Wh_w, nullptr, out, NN);
}